// Classify_Anything_Mixed_OT_Text_Loss_79242146611726
// MI455X (gfx1250) — compile-verified
//
#include <hip/hip_runtime.h>
#include <hip/hip_bf16.h>

#define B_ 4096
#define C_ 8192
#define D_ 512
#define INV_TEMP (1.0f / 0.07f)
#define EPS_ 1e-8f

typedef __attribute__((ext_vector_type(2))) float v2f;
typedef __attribute__((ext_vector_type(8))) float v8f;

// ---------------- workspace layout (floats) ----------------
// simMC  : B_*C_              multiclass sim matrix
// E      : B_*B_              E[i][b] = exp(simMC[b][i])  (transposed exp block)
// rowsum0: B_                 sum_j exp(sim_ij)
// rowmax : B_                 max_j sim_ij
// diag   : B_                 sim_ii
// r      : B_                 row scale
// r2     : B_                 row scale double-buffer
// scal   : 8  -> [0]=m, [1..5]=S per iteration, [6]=mlLoss, [7]=mcLoss
#define WS_SIM    ((size_t)0)
#define WS_E      (WS_SIM + (size_t)B_ * C_)
#define WS_ROWSUM (WS_E + (size_t)B_ * B_)
#define WS_ROWMAX (WS_ROWSUM + B_)
#define WS_DIAG   (WS_ROWMAX + B_)
#define WS_R      (WS_DIAG + B_)
#define WS_R2     (WS_R + B_)
#define WS_SCAL   (WS_R2 + B_)

// ---------------- init: m from dataset_indices, zero accumulators ----------------
__global__ __launch_bounds__(256) void init_kernel(const int* __restrict__ dsi, float* __restrict__ scal) {
    __shared__ float c1[256], c0[256];
    float n1 = 0.f, n0 = 0.f;
    for (int i = threadIdx.x; i < B_; i += 256) {
        int v = dsi[i];
        n1 += (v == 1) ? 1.f : 0.f;
        n0 += (v == 0) ? 1.f : 0.f;
    }
    c1[threadIdx.x] = n1; c0[threadIdx.x] = n0;
    __syncthreads();
    for (int s = 128; s > 0; s >>= 1) {
        if (threadIdx.x < s) { c1[threadIdx.x] += c1[threadIdx.x + s]; c0[threadIdx.x] += c0[threadIdx.x + s]; }
        __syncthreads();
    }
    if (threadIdx.x == 0) {
        scal[0] = c1[0] + 0.1f * c0[0];          // m
        for (int k = 1; k < 8; ++k) scal[k] = 0.f; // S[0..4], mlLoss, mcLoss
    }
}

// ---------------- f32 WMMA 16x16 tile GEMM core ----------------
__device__ __forceinline__ v8f gemm_tile_f32(const float* __restrict__ A,
                                             const float* __restrict__ W,
                                             int row0, int col0, int lane) {
    const int half = lane >> 4;      // 0..1
    const int m16  = lane & 15;      // 0..15
    const float* ap = A + (size_t)(row0 + m16) * D_ + 2 * half;
    const float* bp = W + (size_t)(col0 + m16) * D_ + 2 * half;
    v8f acc = {};
#pragma unroll 8
    for (int k = 0; k < D_; k += 4) {
        v2f a = *(const v2f*)(ap + k);   // A[M=m16][k+2h .. k+2h+1]
        v2f b = *(const v2f*)(bp + k);   // W[N=m16][k+2h .. k+2h+1] (= B^T)
        acc = __builtin_amdgcn_wmma_f32_16x16x4_f32(
            false, a, false, b, (short)0, acc, false, false);
    }
    return acc;
}

// ---------------- GEMM 1: multilabel sim fused with asymmetric loss ----------------
__global__ __launch_bounds__(256) void gemm_ml_kernel(const float* __restrict__ F,
                                                      const float* __restrict__ W,
                                                      const int* __restrict__ tgt,
                                                      float* __restrict__ loss_acc) {
    const int lane = threadIdx.x & 31;
    const int wave = threadIdx.x >> 5;
    const int col0 = blockIdx.x * 128 + wave * 16;
    const int row0 = blockIdx.y * 16;
    v8f acc = gemm_tile_f32(F, W, row0, col0, lane);

    const int half = lane >> 4;
    const int m16  = lane & 15;
    float local = 0.f;
#pragma unroll
    for (int v = 0; v < 8; ++v) {
        const int row = row0 + v + 8 * half;
        const int col = col0 + m16;
        const float x  = acc[v] * INV_TEMP;
        const float yf = (float)tgt[(size_t)row * C_ + col];
        const float xsp  = 1.f / (1.f + expf(-x));
        const float xsn  = fminf(1.f - xsp + 0.05f, 1.f);
        const float base = yf * logf(fmaxf(xsp, EPS_)) + (1.f - yf) * logf(fmaxf(xsn, EPS_));
        const float wb   = fmaxf(1.f - xsp * yf - xsn * (1.f - yf), 0.f);
        const float wp   = powf(wb, yf + 4.f * (1.f - yf));
        local -= base * wp;
    }
    __shared__ float red[256];
    red[threadIdx.x] = local;
    __syncthreads();
    for (int s = 128; s > 0; s >>= 1) {
        if (threadIdx.x < s) red[threadIdx.x] += red[threadIdx.x + s];
        __syncthreads();
    }
    if (threadIdx.x == 0) atomicAdd(loss_acc, red[0]);
}

// ---------------- GEMM 2: multiclass sim -> ws; exp-transposed block -> E ----------------
__global__ __launch_bounds__(256) void gemm_mc_kernel(const float* __restrict__ F,
                                                      const float* __restrict__ W,
                                                      float* __restrict__ sim,
                                                      float* __restrict__ E) {
    const int lane = threadIdx.x & 31;
    const int wave = threadIdx.x >> 5;
    const int col0 = blockIdx.x * 128 + wave * 16;
    const int row0 = blockIdx.y * 16;
    v8f acc = gemm_tile_f32(F, W, row0, col0, lane);

    const int half = lane >> 4;
    const int m16  = lane & 15;
    const bool inblk = (col0 < B_);
#pragma unroll
    for (int v = 0; v < 8; ++v) {
        const int row = row0 + v + 8 * half;
        const int col = col0 + m16;
        const float x = acc[v] * INV_TEMP;
        sim[(size_t)row * C_ + col] = x;
        if (inblk) E[(size_t)col * B_ + row] = expf(x);   // E[i][b] = exp(sim[b][i])
    }
}

// ---------------- per-row stats: rowmax, rowsum0 = sum exp, diag, r=1 ----------------
__global__ __launch_bounds__(256) void row_stats_kernel(const float* __restrict__ sim,
                                                        float* __restrict__ rowsum,
                                                        float* __restrict__ rowmax,
                                                        float* __restrict__ diag,
                                                        float* __restrict__ r) {
    const int i = blockIdx.x;
    const float* rp = sim + (size_t)i * C_;
    float mx = -3.4e38f, sm = 0.f;
    for (int j = threadIdx.x; j < C_; j += 256) {
        const float s = rp[j];
        mx = fmaxf(mx, s);
        sm += expf(s);
    }
    __shared__ float rmx[256], rsm[256];
    rmx[threadIdx.x] = mx; rsm[threadIdx.x] = sm;
    __syncthreads();
    for (int s = 128; s > 0; s >>= 1) {
        if (threadIdx.x < s) {
            rmx[threadIdx.x] = fmaxf(rmx[threadIdx.x], rmx[threadIdx.x + s]);
            rsm[threadIdx.x] += rsm[threadIdx.x + s];
        }
        __syncthreads();
    }
    if (threadIdx.x == 0) {
        rowmax[i] = rmx[0];
        rowsum[i] = rsm[0];
        diag[i]   = rp[i];
        r[i]      = 1.f;
    }
}

// ---------------- OT iteration step A: colsum dot, d-scale, S partial ----------------
__global__ __launch_bounds__(256) void iter_a_kernel(const float* __restrict__ E,
                                                     const float* __restrict__ r,
                                                     const float* __restrict__ diag,
                                                     const float* __restrict__ rowsum,
                                                     float* __restrict__ r2,
                                                     float* __restrict__ scal,
                                                     int it) {
    const int i = blockIdx.x;
    const float* ep = E + (size_t)i * B_;
    float cs = 0.f;
    for (int b = threadIdx.x; b < B_; b += 256) cs += ep[b] * r[b];
    __shared__ float red[256];
    red[threadIdx.x] = cs;
    __syncthreads();
    for (int s = 128; s > 0; s >>= 1) {
        if (threadIdx.x < s) red[threadIdx.x] += red[threadIdx.x + s];
        __syncthreads();
    }
    if (threadIdx.x == 0) {
        const float d  = fminf(diag[i] / red[0], 1.f);
        const float rn = r[i] * d;
        r2[i] = rn;
        atomicAdd(&scal[1 + it], rn * rowsum[i]);   // S partial
    }
}

// ---------------- OT iteration step B: global rescale r = r2 * m/S ----------------
__global__ __launch_bounds__(256) void iter_b_kernel(const float* __restrict__ r2,
                                                     float* __restrict__ r,
                                                     const float* __restrict__ scal,
                                                     int it) {
    const int i = blockIdx.x * 256 + threadIdx.x;
    const float g = scal[0] / scal[1 + it];         // m / S
    if (i < B_) r[i] = r2[i] * g;
}

// ---------------- multiclass CE loss: log_softmax(P) with P = r_i * exp(sim) ----------------
__global__ __launch_bounds__(256) void final_loss_kernel(const float* __restrict__ sim,
                                                         const int* __restrict__ tgt,
                                                         const float* __restrict__ r,
                                                         const float* __restrict__ rowmax,
                                                         float* __restrict__ scal) {
    const int i = blockIdx.x;
    const float* rp = sim + (size_t)i * C_;
    const int*   tp = tgt + (size_t)i * C_;
    const float  ri = r[i];
    const float  Mi = ri * expf(rowmax[i]);         // max_j P_ij
    float se = 0.f, ts = 0.f, tpw = 0.f;
    for (int j = threadIdx.x; j < C_; j += 256) {
        const float P = ri * expf(rp[j]);
        se += expf(P - Mi);
        const float y = (float)tp[j];
        ts  += y;
        tpw += y * P;
    }
    __shared__ float r0[256], r1[256], r2s[256];
    r0[threadIdx.x] = se; r1[threadIdx.x] = ts; r2s[threadIdx.x] = tpw;
    __syncthreads();
    for (int s = 128; s > 0; s >>= 1) {
        if (threadIdx.x < s) {
            r0[threadIdx.x]  += r0[threadIdx.x + s];
            r1[threadIdx.x]  += r1[threadIdx.x + s];
            r2s[threadIdx.x] += r2s[threadIdx.x + s];
        }
        __syncthreads();
    }
    if (threadIdx.x == 0) {
        const float lse    = Mi + logf(r0[0]);
        const float loss_i = r1[0] * lse - r2s[0];  // sum_j y*(lse - P)
        atomicAdd(&scal[7], loss_i * (1.0f / (float)B_));
    }
}

// ---------------- finalize ----------------
__global__ void finalize_kernel(const float* __restrict__ scal, float* __restrict__ out) {
    const float ml = scal[6];
    const float mc = scal[7];
    out[0] = mc + ml;
    out[1] = mc;
    out[2] = ml;
}

extern "C" void kernel_launch(void* const* d_in, const int* in_sizes, int n_in,
                              void* d_out, int out_size, void* d_ws, size_t ws_size,
                              hipStream_t stream) {
    const float* F   = (const float*)d_in[0];   // [B,D]
    const float* Tf  = (const float*)d_in[1];   // [C,D] text_features
    const int*   tgt = (const int*)d_in[2];     // [B,C]
    const int*   dsi = (const int*)d_in[3];     // [B]
    const float* Mc  = (const float*)d_in[4];   // [C,D] multiclass_text_features
    float* out = (float*)d_out;
    float* ws  = (float*)d_ws;

    float* sim    = ws + WS_SIM;
    float* E      = ws + WS_E;
    float* rowsum = ws + WS_ROWSUM;
    float* rowmax = ws + WS_ROWMAX;
    float* diag   = ws + WS_DIAG;
    float* r      = ws + WS_R;
    float* r2     = ws + WS_R2;
    float* scal   = ws + WS_SCAL;

    init_kernel<<<1, 256, 0, stream>>>(dsi, scal);

    const dim3 ggrid(C_ / 128, B_ / 16);
    gemm_ml_kernel<<<ggrid, 256, 0, stream>>>(F, Tf, tgt, &scal[6]);
    gemm_mc_kernel<<<ggrid, 256, 0, stream>>>(F, Mc, sim, E);

    row_stats_kernel<<<B_, 256, 0, stream>>>(sim, rowsum, rowmax, diag, r);

    for (int it = 0; it < 5; ++it) {
        iter_a_kernel<<<B_, 256, 0, stream>>>(E, r, diag, rowsum, r2, scal, it);
        iter_b_kernel<<<B_ / 256, 256, 0, stream>>>(r2, r, scal, it);
    }

    final_loss_kernel<<<B_, 256, 0, stream>>>(sim, tgt, r, rowmax, scal);
    finalize_kernel<<<1, 1, 0, stream>>>(scal, out);
}